// CTCCriterion_2035814498851
// MI455X (gfx1250) — compile-verified
//
#include <hip/hip_runtime.h>
#include <stdint.h>

// Problem sizes are fixed by the reference harness.
#define CTC_B 64
#define CTC_T 1024
#define CTC_V 512
#define CTC_S 128
#define CTC_L (2 * CTC_S + 1)   // 257 extended states
#define NEGF (-1e30f)

// ---- CDNA5 async global->LDS path (ASYNCcnt-tracked) -----------------------
#if defined(__has_builtin)
#if __has_builtin(__builtin_amdgcn_global_load_async_to_lds_b128) && \
    __has_builtin(__builtin_amdgcn_s_wait_asynccnt)
#define HAVE_ASYNC_LDS 1
#endif
#endif

#if defined(HAVE_ASYNC_LDS)
typedef int v4i_t __attribute__((ext_vector_type(4)));
typedef __attribute__((address_space(1))) v4i_t* gas_v4i_p;  // global (AS1)
typedef __attribute__((address_space(3))) v4i_t* las_v4i_p;  // LDS    (AS3)
#endif

// ---------------------------------------------------------------------------
// Kernel 1: one block per (b,t) row.
//   lse = logsumexp(logits[b,t,:])
//   lp_blank[b*T+t]      = logits[b,t,0]            - lse
//   lp_label[b*T+t][j]   = logits[b,t,target[b,j]]  - lse   (j = 0..S-1)
// ---------------------------------------------------------------------------
__global__ __launch_bounds__(128) void ctc_row_kernel(
    const float* __restrict__ logits,   // [B,T,V]
    const int*   __restrict__ targets,  // [B,S]
    float*       __restrict__ lp_blank, // [B*T]
    float*       __restrict__ lp_label) // [B*T, S]
{
  __shared__ float row[CTC_V];
  __shared__ float red_max[4];
  __shared__ float red_sum[4];

  const int rowid = blockIdx.x;        // b*T + t
  const int b     = rowid >> 10;       // T == 1024
  const int tid   = threadIdx.x;       // 0..127
  const int lane  = tid & 31;
  const int wave  = tid >> 5;
  const float* src = logits + (size_t)rowid * CTC_V;

#if defined(HAVE_ASYNC_LDS)
  // 128 lanes x 16B = 2KB row staged straight into LDS, bypassing VGPRs
  // (GLOBAL_LOAD_ASYNC_TO_LDS_B128, completion via ASYNCcnt).
  // Address-space casts go through uintptr_t: AS1 bits == generic bits for
  // global memory; AS3 offset == low 32 bits of the generic LDS address.
  {
    gas_v4i_p gsrc = (gas_v4i_p)(uintptr_t)(src + tid * 4);
    las_v4i_p ldst = (las_v4i_p)(uint32_t)(uintptr_t)(&row[tid * 4]);
    __builtin_amdgcn_global_load_async_to_lds_b128(gsrc, ldst, 0, 0);
    __builtin_amdgcn_s_wait_asynccnt(0);
  }
#else
  const float4 v = ((const float4*)src)[tid];
  ((float4*)row)[tid] = v;
#endif
  __syncthreads();

  const float x0 = row[tid];
  const float x1 = row[tid + 128];
  const float x2 = row[tid + 256];
  const float x3 = row[tid + 384];

  // max over V (wave32 shuffle + 4-wave LDS combine)
  float m = fmaxf(fmaxf(x0, x1), fmaxf(x2, x3));
  #pragma unroll
  for (int off = 16; off > 0; off >>= 1) m = fmaxf(m, __shfl_xor(m, off, 32));
  if (lane == 0) red_max[wave] = m;
  __syncthreads();
  m = fmaxf(fmaxf(red_max[0], red_max[1]), fmaxf(red_max[2], red_max[3]));

  // sum exp(x - m)
  float s = __expf(x0 - m) + __expf(x1 - m) + __expf(x2 - m) + __expf(x3 - m);
  #pragma unroll
  for (int off = 16; off > 0; off >>= 1) s += __shfl_xor(s, off, 32);
  if (lane == 0) red_sum[wave] = s;
  __syncthreads();
  const float lse = m + __logf(red_sum[0] + red_sum[1] + red_sum[2] + red_sum[3]);

  // gather per-state log-probs (labels are in [1,V), blank = 0)
  const int lbl = targets[b * CTC_S + tid];           // tid == j, S == blockDim
  lp_label[(size_t)rowid * CTC_S + tid] = row[lbl] - lse;
  if (tid == 0) lp_blank[rowid] = row[0] - lse;
}

// ---------------------------------------------------------------------------
// Kernel 2: alpha recursion. One block per batch, one thread per state,
// LDS double buffer with a 2-slot NEG pad so a[s-1], a[s-2] need no branches.
// ---------------------------------------------------------------------------
__global__ __launch_bounds__(288) void ctc_alpha_kernel(
    const float* __restrict__ lp_blank_all, // [B*T]
    const float* __restrict__ lp_label_all, // [B*T, S]
    const int*   __restrict__ targets,      // [B,S]
    const int*   __restrict__ in_len,       // [B]
    const int*   __restrict__ tgt_len,      // [B]
    float*       __restrict__ loss_b)       // [B]
{
  __shared__ float abuf[2][CTC_L + 2];      // [parity][2 NEG pads + 257 states]

  const int b = blockIdx.x;
  const int s = threadIdx.x;                // 0..287 (257 active)
  const bool active = (s < CTC_L);
  if (s < 2) { abuf[0][s] = NEGF; abuf[1][s] = NEGF; }

  const bool odd = (s & 1) != 0;
  const int  j   = odd ? ((s - 1) >> 1) : 0;
  bool can_skip = false;
  if (active && odd && s >= 3) {            // skip iff label differs from label 2 back
    can_skip = targets[b * CTC_S + j] != targets[b * CTC_S + j - 1];
  }

  const float* lpb = lp_blank_all + (size_t)b * CTC_T;
  const float* lpl = lp_label_all + (size_t)b * CTC_T * CTC_S;
  const int len = in_len[b];

  // t = 0 init
  float a_self = NEGF;
  if (active) {
    if (s == 0)      a_self = lpb[0];
    else if (s == 1) a_self = lpl[0];
    abuf[0][s + 2] = a_self;
  }
  __syncthreads();

  // software prefetch of lp for t = 1 (next-step loads hide behind the barrier)
  float lp_next = 0.0f;
  if (active) lp_next = odd ? lpl[(size_t)CTC_S + j] : lpb[1];

  int p = 0;
  for (int t = 1; t < CTC_T; ++t) {
    if (t >= len) break;                    // uniform per block; alpha frozen after len
    const float lp_cur = lp_next;
    if (active && (t + 1 < CTC_T))
      lp_next = odd ? lpl[(size_t)(t + 1) * CTC_S + j] : lpb[t + 1];

    if (active) {
      const float a1 = abuf[p][s + 1];                       // alpha[s-1]
      const float a2 = can_skip ? abuf[p][s] : NEGF;         // alpha[s-2]
      const float mm  = fmaxf(a_self, fmaxf(a1, a2));
      const float sum = __expf(a_self - mm) + __expf(a1 - mm) + __expf(a2 - mm);
      a_self = mm + __logf(sum) + lp_cur;
      abuf[p ^ 1][s + 2] = a_self;
    }
    __syncthreads();
    p ^= 1;
  }

  if (s == 0) {
    const int tl = tgt_len[b];
    const int e  = 2 * tl;                                   // terminal states e, e-1
    const float x = abuf[p][e + 2];
    const float y = abuf[p][e + 1];
    const float mm = fmaxf(x, y);
    const float ll = mm + __logf(__expf(x - mm) + __expf(y - mm));
    float lo = -ll;
    if (lo > 1e29f) lo = 0.0f;                               // zero_infinity
    loss_b[b] = lo;
  }
}

// ---------------------------------------------------------------------------
// Kernel 3: sum the 64 per-batch losses into d_out[0].
// ---------------------------------------------------------------------------
__global__ __launch_bounds__(64) void ctc_sum_kernel(
    const float* __restrict__ loss_b, float* __restrict__ out)
{
  const int tid = threadIdx.x;              // 0..63
  float v = loss_b[tid];
  #pragma unroll
  for (int off = 16; off > 0; off >>= 1) v += __shfl_xor(v, off, 32);
  __shared__ float red[2];
  if ((tid & 31) == 0) red[tid >> 5] = v;
  __syncthreads();
  if (tid == 0) out[0] = red[0] + red[1];
}

// ---------------------------------------------------------------------------
extern "C" void kernel_launch(void* const* d_in, const int* in_sizes, int n_in,
                              void* d_out, int out_size, void* d_ws, size_t ws_size,
                              hipStream_t stream) {
  (void)in_sizes; (void)n_in; (void)out_size; (void)ws_size;

  const float* logits  = (const float*)d_in[0];  // [B,T,V] fp32
  const int*   out_len = (const int*)  d_in[1];  // [B]
  const int*   targets = (const int*)  d_in[2];  // [B,S]
  const int*   tgt_len = (const int*)  d_in[3];  // [B]

  // workspace layout (floats): lp_blank [B*T] | lp_label [B*T*S] | loss_b [B]
  float* ws       = (float*)d_ws;
  float* lp_blank = ws;
  float* lp_label = ws + (size_t)CTC_B * CTC_T;
  float* loss_b   = lp_label + (size_t)CTC_B * CTC_T * CTC_S;

  ctc_row_kernel<<<dim3(CTC_B * CTC_T), dim3(128), 0, stream>>>(
      logits, targets, lp_blank, lp_label);
  ctc_alpha_kernel<<<dim3(CTC_B), dim3(288), 0, stream>>>(
      lp_blank, lp_label, targets, out_len, tgt_len, loss_b);
  ctc_sum_kernel<<<dim3(1), dim3(64), 0, stream>>>(loss_b, (float*)d_out);
}